// Gator_4286377361936
// MI455X (gfx1250) — compile-verified
//
#include <hip/hip_runtime.h>

// Logic-gate-network forward pass, bit-plane (SWAR) formulation for gfx1250.
// One wave (32 threads) per block owns 32 batch elements; the growing 9216-col
// boolean state lives bit-transposed in LDS (1 bit per batch element per col).
// Each lane evaluates one gate for all 32 batch elements with pure bit logic.
// Streaming input/output uses NT-hinted vmem; ballot_w32 does the bit transpose.

constexpr int W0   = 1024;            // initial width
constexpr int R    = 8;               // gate rows (layers)
constexpr int G    = 1024;            // gates per row
constexpr int WTOT = W0 + R * G;      // 9216 final columns
constexpr int BPB  = 32;              // batch elements per block (== wave32)

typedef float  vfloat4 __attribute__((ext_vector_type(4)));   // native vector for NT builtins

__global__ __launch_bounds__(32) void lgn_bitplane_kernel(
    const float* __restrict__ x,        // [B, W0] 0/1 floats
    const float* __restrict__ gates,    // [R, G, 4] 0/1 floats
    const int*   __restrict__ choices,  // [R, G, 2] int32
    float* __restrict__ out)            // [B, WTOT] 0/1 floats
{
    __shared__ unsigned      sh_bp[WTOT];      // bitplane: word j = column j over 32 batch elems
    __shared__ unsigned char sh_lut[R * G];    // 4-bit truth tables, one byte each

    const int lane = threadIdx.x;              // 0..31 (wave32)
    const int b0   = blockIdx.x * BPB;

    // ---- Phase 1: pack truth tables (4 floats) into nibbles in LDS ----
    const float4* g4 = (const float4*)gates;
    for (int i = lane; i < R * G; i += 32) {
        float4 t = g4[i];
        sh_lut[i] = (unsigned char)((t.x != 0.0f ? 1u : 0u) |
                                    (t.y != 0.0f ? 2u : 0u) |
                                    (t.z != 0.0f ? 4u : 0u) |
                                    (t.w != 0.0f ? 8u : 0u));
    }

    // ---- Phase 2: coalesced NT input reads + wave32 ballot bit-transpose ----
    // For each 32-column chunk: gather 32 row-major words (bits = columns) via
    // ballot, transpose to 32 bitplane words (bits = batch) via ballot; each
    // lane keeps its own transposed word -> one conflict-free ds_store/chunk.
    for (int w = 0; w < W0 / 32; ++w) {
        unsigned rowWord = 0;
#pragma unroll
        for (int bl = 0; bl < 32; ++bl) {      // 32 independent 128B-coalesced loads
            float v = __builtin_nontemporal_load(
                x + (size_t)(b0 + bl) * W0 + w * 32 + lane);
            unsigned m = __builtin_amdgcn_ballot_w32(v != 0.0f);
            rowWord = (lane == bl) ? m : rowWord;      // branchless v_cndmask
        }
        unsigned myWord = 0;
#pragma unroll
        for (int j = 0; j < 32; ++j) {
            unsigned mj = __builtin_amdgcn_ballot_w32(((rowWord >> j) & 1u) != 0u);
            myWord = (lane == j) ? mj : myWord;        // branchless v_cndmask
        }
        sh_bp[w * 32 + lane] = myWord;                 // consecutive -> conflict-free
    }
    __syncthreads();

    // ---- Phase 3: evaluate gate rows; lane = 1 gate x 32 batch (bitwise) ----
    const int2* ch2 = (const int2*)choices;
    for (int r = 0; r < R; ++r) {
        // Preload this lane's 32 input-pair descriptors for the row (L2-resident,
        // coalesced b64 loads, all issued before first use).
        int2 cs[G / 32];
#pragma unroll
        for (int it = 0; it < G / 32; ++it)
            cs[it] = ch2[r * G + it * 32 + lane];

        const unsigned char* lutRow = &sh_lut[r * G];
        const int outBase = W0 + r * G;        // reads are guaranteed < outBase
#pragma unroll 4
        for (int it = 0; it < G / 32; ++it) {  // 4 independent DS chains in flight
            const int g = it * 32 + lane;
            const int2 c = cs[it];
            const unsigned a = sh_bp[c.x];     // 32 batch elems of input 0
            const unsigned b = sh_bp[c.y];     // 32 batch elems of input 1
            const unsigned lut = lutRow[g];
            const unsigned m0 = (lut & 1u) ? 0xffffffffu : 0u;  // a=0,b=0
            const unsigned m1 = (lut & 2u) ? 0xffffffffu : 0u;  // a=0,b=1
            const unsigned m2 = (lut & 4u) ? 0xffffffffu : 0u;  // a=1,b=0
            const unsigned m3 = (lut & 8u) ? 0xffffffffu : 0u;  // a=1,b=1
            const unsigned res = (m3 & a & b) | (m2 & a & ~b) |
                                 (m1 & ~a & b) | (m0 & ~a & ~b);
            sh_bp[outBase + g] = res;          // consecutive words -> conflict-free
        }
        __syncthreads();
    }

    // ---- Phase 4: expand bitplanes to 0/1 floats, NT-hinted b128 stores ----
    const uint4* bp4 = (const uint4*)sh_bp;
    for (int jq = 0; jq < WTOT / 128; ++jq) {            // 72 chunks of 128 cols
        uint4 wv = bp4[jq * 32 + lane];                  // lane covers 4 columns
        float* ob = out + (size_t)b0 * WTOT + (size_t)jq * 128 + lane * 4;
#pragma unroll
        for (int i = 0; i < 32; ++i) {                   // batch element i
            vfloat4 v;
            v.x = ((wv.x >> i) & 1u) ? 1.0f : 0.0f;
            v.y = ((wv.y >> i) & 1u) ? 1.0f : 0.0f;
            v.z = ((wv.z >> i) & 1u) ? 1.0f : 0.0f;
            v.w = ((wv.w >> i) & 1u) ? 1.0f : 0.0f;
            __builtin_nontemporal_store(v, (vfloat4*)(ob + (size_t)i * WTOT));
        }
    }
}

extern "C" void kernel_launch(void* const* d_in, const int* in_sizes, int n_in,
                              void* d_out, int out_size, void* d_ws, size_t ws_size,
                              hipStream_t stream) {
    const float* x       = (const float*)d_in[0];   // [B, W0] float32
    const float* gates   = (const float*)d_in[1];   // [R, G, 4] float32
    const int*   choices = (const int*)d_in[2];     // [R, G, 2] int32
    float*       out     = (float*)d_out;           // [B, WTOT] float32

    const int B = in_sizes[0] / W0;                 // 8192
    const int nblocks = B / BPB;                    // 256 single-wave blocks
    lgn_bitplane_kernel<<<nblocks, 32, 0, stream>>>(x, gates, choices, out);
    (void)d_ws; (void)ws_size; (void)out_size; (void)n_in;
}